// OneDirectionChamferDistanceWithIdx_87222195847732
// MI455X (gfx1250) — compile-verified
//
#include <hip/hip_runtime.h>

typedef __attribute__((ext_vector_type(2))) float v2f;
typedef __attribute__((ext_vector_type(8))) float v8f;

#define BATCH   2
#define NPTS    8192
#define MPTS    8192
#define MTILES  (MPTS / 16)      // 512
#define WAVES   2                // waves per block
#define BLOCK   (WAVES * 32)

// ---------------------------------------------------------------------------
// Pack xyz2 into the exact VGPR layout of the f32 4x16 B matrix for
// V_WMMA_F32_16X16X4_F32:  per 16-col tile, lane n (0-15) supplies rows
// K=0,1 = (y0, y1); lane 16+n supplies rows K=2,3 = (y2, |y|^2).
// One float2 (b64) load per lane per tile in the main loop.
// ---------------------------------------------------------------------------
__global__ void pack_b_kernel(const float* __restrict__ xyz2,
                              float2* __restrict__ pb) {
    int gid = blockIdx.x * blockDim.x + threadIdx.x;
    if (gid >= BATCH * MPTS) return;
    int b = gid / MPTS;
    int p = gid - b * MPTS;
    const float* y = xyz2 + ((size_t)b * MPTS + p) * 3;
    float y0 = y[0], y1 = y[1], y2 = y[2];
    int tile = p >> 4, r = p & 15;
    size_t base = ((size_t)b * MTILES + tile) * 32;
    pb[base + r]      = make_float2(y0, y1);
    pb[base + 16 + r] = make_float2(y2, y0 * y0 + y1 * y1 + y2 * y2);
}

// ---------------------------------------------------------------------------
// Main kernel: each wave owns a 16-row strip of xyz1 for one batch and sweeps
// all 512 column tiles of xyz2 with one f32 WMMA per tile:
//   A row i = (-2x0, -2x1, -2x2, 1),  B col j = (y0, y1, y2, |y|^2)
//   D[i][j] = |y_j|^2 - 2 x_i . y_j     (same argmin as the true distance)
// Argmin tracked per (vgpr, lane) class; reduced across the 16 lanes of each
// row half via LDS; exact distance recomputed once per row for the winner.
// ---------------------------------------------------------------------------
__global__ __launch_bounds__(BLOCK)
void chamfer_min_kernel(const float* __restrict__ xyz1,
                        const float* __restrict__ xyz2,
                        const float2* __restrict__ pb,
                        float* __restrict__ dist_out,
                        int* __restrict__ idx_out) {
    __shared__ float sval[WAVES][8][32];
    __shared__ int   sidx[WAVES][8][32];

    const int lane  = threadIdx.x & 31;
    const int wv    = threadIdx.x >> 5;
    const int strip = blockIdx.x * WAVES + wv;
    const int b     = strip / (NPTS / 16);
    const int i0    = (strip - b * (NPTS / 16)) * 16;

    // ---- A operand: 16x4 f32, lanes 0-15 -> K=0,1 ; lanes 16-31 -> K=2,3
    const int m = lane & 15;
    const float* xp = xyz1 + ((size_t)b * NPTS + i0 + m) * 3;
    const float x0 = xp[0], x1 = xp[1], x2 = xp[2];
    v2f a;
    if (lane < 16) { a.x = -2.0f * x0; a.y = -2.0f * x1; }
    else           { a.x = -2.0f * x2; a.y = 1.0f;       }

    float minv[8];
    int   mini[8];
#pragma unroll
    for (int v = 0; v < 8; ++v) { minv[v] = __builtin_inff(); mini[v] = 0; }

    const float2* pbb = pb + ((size_t)b * MTILES) * 32 + lane;
    const int jcol = lane & 15;

#pragma unroll 2
    for (int t = 0; t < MTILES; ++t) {
        float2 bv = pbb[(size_t)t * 32];          // global_load_b64, streams L2/WGP$
        v2f bop; bop.x = bv.x; bop.y = bv.y;
        v8f c = {};
        v8f d = __builtin_amdgcn_wmma_f32_16x16x4_f32(
            /*neg_a=*/false, a, /*neg_b=*/false, bop,
            /*c_mod=*/(short)0, c, /*reuse_a=*/false, /*reuse_b=*/false);

        const int j = (t << 4) + jcol;            // column index for this lane
#pragma unroll
        for (int v = 0; v < 8; ++v) {
            bool lt = d[v] < minv[v];             // strict < keeps lowest j in class
            mini[v] = lt ? j    : mini[v];
            minv[v] = lt ? d[v] : minv[v];
        }
    }

    // ---- reduce 16 lanes per row (row m -> vgpr m%8, lane half m/8)
#pragma unroll
    for (int v = 0; v < 8; ++v) {
        sval[wv][v][lane] = minv[v];
        sidx[wv][v][lane] = mini[v];
    }
    __syncthreads();

    if (lane < 16) {
        const int half = lane >> 3;               // row m = lane: vgpr = m%8, half = m/8
        const int v    = lane & 7;
        float bestv = sval[wv][v][half * 16];
        int   besti = sidx[wv][v][half * 16];
#pragma unroll
        for (int n = 1; n < 16; ++n) {
            float cv = sval[wv][v][half * 16 + n];
            int   ci = sidx[wv][v][half * 16 + n];
            if (cv < bestv || (cv == bestv && ci < besti)) { bestv = cv; besti = ci; }
        }
        // exact distance for the winner (reference-accurate, no cancellation)
        const int i = i0 + lane;
        const float* yp = xyz2 + ((size_t)b * MPTS + besti) * 3;
        float dx = x0 - yp[0], dy = x1 - yp[1], dz = x2 - yp[2];
        dist_out[(size_t)b * NPTS + i] = dx * dx + dy * dy + dz * dz;
        idx_out [(size_t)b * NPTS + i] = besti;
    }
}

// ---------------------------------------------------------------------------
extern "C" void kernel_launch(void* const* d_in, const int* in_sizes, int n_in,
                              void* d_out, int out_size, void* d_ws, size_t ws_size,
                              hipStream_t stream) {
    const float* xyz1 = (const float*)d_in[0];   // [B, N, 3] f32
    const float* xyz2 = (const float*)d_in[1];   // [B, M, 3] f32
    float2* pb = (float2*)d_ws;                  // 256 KB packed B operand

    float* dist = (float*)d_out;                 // dist1: [B, N] f32
    int*   idx  = ((int*)d_out) + BATCH * NPTS;  // idx1:  [B, N] i32 (concat)

    pack_b_kernel<<<(BATCH * MPTS + 255) / 256, 256, 0, stream>>>(xyz2, pb);

    const int nstrips = BATCH * (NPTS / 16);     // 1024 waves
    chamfer_min_kernel<<<nstrips / WAVES, BLOCK, 0, stream>>>(
        xyz1, xyz2, pb, dist, idx);
}